// Conv2d_54881092108704
// MI455X (gfx1250) — compile-verified
//
#include <hip/hip_runtime.h>

typedef float v2f __attribute__((ext_vector_type(2)));
typedef float v8f __attribute__((ext_vector_type(8)));

#define NBATCH 32
#define C_IN   64
#define H_IN   224
#define W_IN   224
#define OC     128
#define H_OUT  222
#define W_OUT  222
#define PIX    (H_IN * W_IN)     // 50176
#define OPIX   (H_OUT * W_OUT)   // 49284
#define NCTILE 14                // ceil(222/16)

// ---------------------------------------------------------------------------
// Kernel 1: xs[b, p] = sum_c x[b, c, p]   (411 MB streamed read, 6.4 MB write)
// ---------------------------------------------------------------------------
__global__ void __launch_bounds__(256)
chansum_kernel(const float* __restrict__ x, float* __restrict__ xs) {
    int idx = blockIdx.x * 256 + threadIdx.x;      // [0, 32*50176)
    int b = idx / PIX;
    int p = idx - b * PIX;
    const float* px = x + (size_t)b * C_IN * PIX + p;
    float s = 0.0f;
#pragma unroll 8
    for (int c = 0; c < C_IN; ++c) {
        s += __builtin_nontemporal_load(px + (size_t)c * PIX);  // NT: streamed
    }
    xs[idx] = s;   // regular store: keep xs resident in L2 for kernel 2
}

// ---------------------------------------------------------------------------
// Kernel 2: 3x3 VALID conv of xs against 128 filters via V_WMMA_F32_16X16X4_F32
// One wave -> 16 consecutive output pixels x all 128 output channels.
// Taps 0..8 padded to K=12, processed as three K=4 WMMA groups.
// ---------------------------------------------------------------------------
__global__ void __launch_bounds__(256)
conv_wmma_kernel(const float* __restrict__ xs, const float* __restrict__ w,
                 float* __restrict__ out) {
    __shared__ float sW[OC * 9];                   // 4.6 KB filter stage
    int tid = threadIdx.x;
    for (int i = tid; i < OC * 9; i += 256) sW[i] = w[i];
    __syncthreads();

    int wave = tid >> 5;
    int lane = tid & 31;
    int half = lane >> 4;                          // selects K pair {0,1} vs {2,3}
    int n    = lane & 15;                          // N (pixel) for B/D, M (oc) for A

    int tile = blockIdx.x * 8 + wave;              // [0, 32*222*14)
    int ct   = tile % NCTILE;
    int t2   = tile / NCTILE;
    int oh   = t2 % H_OUT;
    int b    = t2 / H_OUT;
    int ow   = ct * 16 + n;
    int owc  = ow < W_OUT ? ow : (W_OUT - 1);      // clamp loads for edge tile

    // B matrix (4x16 per group): lane-half picks K={0,1} or {2,3}; vgpr j steps K.
    const float* xrow = xs + (size_t)b * PIX;
    v2f Bg[3];
#pragma unroll
    for (int g = 0; g < 3; ++g) {
#pragma unroll
        for (int j = 0; j < 2; ++j) {
            int t = 4 * g + 2 * half + j;          // tap index 0..11
            float v = 0.0f;
            if (t <= 8) {
                int kh = t / 3, kw = t - kh * 3;
                v = xrow[(oh + kh) * W_IN + owc + kw];
            }
            Bg[g][j] = v;
        }
    }

    // 8 oc-tiles of 16 channels; A matrix 16x4 from LDS (same lane-half K rule).
#pragma unroll
    for (int ot = 0; ot < 8; ++ot) {
        v8f acc = {};
#pragma unroll
        for (int g = 0; g < 3; ++g) {
            v2f A;
#pragma unroll
            for (int j = 0; j < 2; ++j) {
                int t = 4 * g + 2 * half + j;
                A[j] = (t <= 8) ? sW[(ot * 16 + n) * 9 + t] : 0.0f;
            }
            // D = A(16x4) * B(4x16) + C : v_wmma_f32_16x16x4_f32
            acc = __builtin_amdgcn_wmma_f32_16x16x4_f32(
                /*neg_a=*/false, A, /*neg_b=*/false, Bg[g],
                /*c_mod=*/(short)0, acc, /*reuse_a=*/false, /*reuse_b=*/false);
        }
        // D layout: vgpr r = row M=r (lanes 0-15) / M=8+r (lanes 16-31), N=lane%16.
        if (ow < W_OUT) {
            size_t base = (((size_t)b * OC + ot * 16 + half * 8) * (size_t)OPIX)
                        + (size_t)oh * W_OUT + ow;
#pragma unroll
            for (int r = 0; r < 8; ++r) {
                // NT store: 807 MB streamed output must not thrash xs out of L2
                __builtin_nontemporal_store(acc[r], out + base + (size_t)r * OPIX);
            }
        }
    }
}

// ---------------------------------------------------------------------------
extern "C" void kernel_launch(void* const* d_in, const int* in_sizes, int n_in,
                              void* d_out, int out_size, void* d_ws, size_t ws_size,
                              hipStream_t stream) {
    const float* x = (const float*)d_in[0];   // (32, 64, 224, 224) f32
    const float* w = (const float*)d_in[1];   // (128, 3, 3) f32
    float* out = (float*)d_out;               // (32, 128, 222, 222) f32
    float* xs  = (float*)d_ws;                // scratch: 32*224*224 f32 = 6.1 MiB

    chansum_kernel<<<(NBATCH * PIX) / 256, 256, 0, stream>>>(x, xs);

    int ntiles = NBATCH * H_OUT * NCTILE;     // 99456, divisible by 8 waves/block
    conv_wmma_kernel<<<ntiles / 8, 256, 0, stream>>>(xs, w, out);
}